// HGAT_sparse_70944269795863
// MI455X (gfx1250) — compile-verified
//
#include <hip/hip_runtime.h>
#include <math.h>

#define N_NODE 20000
#define N_EDGE 5000
#define DEG    40
#define D      128
#define N_PAIR (N_EDGE * DEG)

typedef float v2f __attribute__((ext_vector_type(2)));
typedef float v8f __attribute__((ext_vector_type(8)));

// ---------------------------------------------------------------------------
// C[M x 128] = A[M x 128] @ W[128 x 128], f32, via V_WMMA_F32_16X16X4_F32.
// Block = 256 threads = 8 waves; wave w computes the 16x16 tile at
// (rowTile*16, w*16). K loop fully unrolled -> 32 WMMAs per wave.
// ---------------------------------------------------------------------------
__global__ __launch_bounds__(256) void gemm_wmma_f32(const float* __restrict__ A,
                                                     const float* __restrict__ W,
                                                     float* __restrict__ C, int M) {
    const int lane = threadIdx.x & 31;
    const int wave = threadIdx.x >> 5;     // col tile 0..7
    const int l16  = lane & 15;
    const int hi   = lane >> 4;            // lane group 0/1
    const int rowBase = blockIdx.x * 16;
    const int colBase = wave * 16;

    int arow = rowBase + l16;              // A operand row (M = lane%16)
    if (arow >= M) arow = M - 1;           // clamp; garbage rows never stored
    const float* Arow = A + (size_t)arow * D;

    v8f c = {0.f, 0.f, 0.f, 0.f, 0.f, 0.f, 0.f, 0.f};
#pragma unroll
    for (int k0 = 0; k0 < D; k0 += 4) {
        const int kk = k0 + 2 * hi;        // lane group covers K pair {kk, kk+1}
        v2f a, b;
        a.x = Arow[kk];
        a.y = Arow[kk + 1];
        b.x = W[(size_t)kk * D + colBase + l16];
        b.y = W[(size_t)(kk + 1) * D + colBase + l16];
        c = __builtin_amdgcn_wmma_f32_16x16x4_f32(false, a, false, b,
                                                  (short)0, c, false, false);
    }
#pragma unroll
    for (int v = 0; v < 8; ++v) {
        const int row = rowBase + v + 8 * hi;   // C VGPR v -> row v + 8*group
        if (row < M) C[(size_t)row * D + colBase + l16] = c[v];
    }
}

// ---------------------------------------------------------------------------
// Per-pair attention score: s = sum_d xe'[e][d] * x'[n][d] * a[d];
// es = exp(s); Z_edge[e] += es; Z_node[n] += es.  One wave per pair.
// ---------------------------------------------------------------------------
__global__ __launch_bounds__(256) void pair_scores(const float* __restrict__ xeP,
                                                   const float* __restrict__ xP,
                                                   const int* __restrict__ pe,
                                                   const int* __restrict__ pn,
                                                   const float* __restrict__ a,
                                                   float* __restrict__ es,
                                                   float* __restrict__ Ze,
                                                   float* __restrict__ Zn) {
    const int lane = threadIdx.x & 31;
    const int p = blockIdx.x * 8 + (threadIdx.x >> 5);
    if (p >= N_PAIR) return;
    const int e = pe[p], n = pn[p];
    const float4 u = ((const float4*)(xeP + (size_t)e * D))[lane];
    const float4 v = ((const float4*)(xP  + (size_t)n * D))[lane];
    const float4 w = ((const float4*)a)[lane];
    float s = u.x * v.x * w.x + u.y * v.y * w.y + u.z * v.z * w.z + u.w * v.w * w.w;
#pragma unroll
    for (int off = 16; off > 0; off >>= 1) s += __shfl_xor(s, off, 32);
    if (lane == 0) {
        const float ev = __expf(s);
        es[p] = ev;
        unsafeAtomicAdd(Ze + e, ev);
        unsafeAtomicAdd(Zn + n, ev);
    }
}

// Column sums of an [M x 128] matrix into out[128].
__global__ __launch_bounds__(128) void colsum(const float* __restrict__ A, int M,
                                              float* __restrict__ out) {
    const int d = threadIdx.x;
    const int r0 = blockIdx.x * 128;
    float acc = 0.f;
    for (int r = r0; r < r0 + 128 && r < M; ++r) acc += A[(size_t)r * D + d];
    unsafeAtomicAdd(out + d, acc);
}

// xe_out[e] = elu( (1e-10*colsum_x + sum_i es * x'[n_i]) / (N_NODE*1e-10 + Ze[e]) )
// Pairs are grouped by edge (e*DEG .. e*DEG+39) -> no atomics needed.
__global__ __launch_bounds__(128) void edge_out(const float* __restrict__ xP,
                                                const float* __restrict__ es,
                                                const int* __restrict__ pn,
                                                const float* __restrict__ Ze,
                                                const float* __restrict__ csx,
                                                float* __restrict__ out) {
    const int e = blockIdx.x;
    const int d = threadIdx.x;
    float acc = 1e-10f * csx[d];
    const int base = e * DEG;
#pragma unroll 4
    for (int i = 0; i < DEG; ++i) {
        const int p = base + i;
        acc = fmaf(es[p], xP[(size_t)pn[p] * D + d], acc);
    }
    const float v = acc / ((float)N_NODE * 1e-10f + Ze[e]);
    out[(size_t)e * D + d] = v > 0.f ? v : __expf(v) - 1.f;
}

// Scatter es[p] * xe'[e] into S_node[n] with f32 atomics.
__global__ __launch_bounds__(128) void node_scatter(const float* __restrict__ xeP,
                                                    const float* __restrict__ es,
                                                    const int* __restrict__ pe,
                                                    const int* __restrict__ pn,
                                                    float* __restrict__ Sn) {
    const int p = blockIdx.x;
    const int d = threadIdx.x;
    const int e = pe[p], n = pn[p];
    unsafeAtomicAdd(Sn + (size_t)n * D + d, es[p] * xeP[(size_t)e * D + d]);
}

// x_out[n] = elu( (1e-10*colsum_xe + S_node[n]) / (N_EDGE*1e-10 + Zn[n]) )
__global__ __launch_bounds__(128) void node_out(const float* __restrict__ Sn,
                                                const float* __restrict__ Zn,
                                                const float* __restrict__ csxe,
                                                float* __restrict__ out) {
    const int n = blockIdx.x;
    const int d = threadIdx.x;
    const float v = (1e-10f * csxe[d] + Sn[(size_t)n * D + d]) /
                    ((float)N_EDGE * 1e-10f + Zn[n]);
    out[(size_t)n * D + d] = v > 0.f ? v : __expf(v) - 1.f;
}

extern "C" void kernel_launch(void* const* d_in, const int* in_sizes, int n_in,
                              void* d_out, int out_size, void* d_ws, size_t ws_size,
                              hipStream_t stream) {
    const float* x    = (const float*)d_in[0];   // [N_NODE, D]
    const float* xe   = (const float*)d_in[1];   // [N_EDGE, D]
    const int*   pair = (const int*)d_in[2];     // [2, N_PAIR]
    const float* a    = (const float*)d_in[3];   // [D, 1]
    const float* wt   = (const float*)d_in[4];   // [D, D]
    const int* pe = pair;                        // edge ids
    const int* pn = pair + N_PAIR;               // node ids

    float* out = (float*)d_out;                  // [x_out | xe_out], row-major f32
    float* out_x  = out;                         // N_NODE*D
    float* out_xe = out + (size_t)N_NODE * D;    // N_EDGE*D

    // workspace carve (floats); zeroed region is contiguous [Sn .. csxe]
    float* ws   = (float*)d_ws;
    float* xP   = ws;                                  // N_NODE*D
    float* xeP  = xP  + (size_t)N_NODE * D;            // N_EDGE*D
    float* es   = xeP + (size_t)N_EDGE * D;            // N_PAIR
    float* Sn   = es  + (size_t)N_PAIR;                // N_NODE*D   (zeroed)
    float* Ze   = Sn  + (size_t)N_NODE * D;            // N_EDGE     (zeroed)
    float* Zn   = Ze  + N_EDGE;                        // N_NODE     (zeroed)
    float* csx  = Zn  + N_NODE;                        // D          (zeroed)
    float* csxe = csx + D;                             // D          (zeroed)
    const size_t zeroFloats = (size_t)N_NODE * D + N_EDGE + N_NODE + 2 * D;
    hipMemsetAsync(Sn, 0, zeroFloats * sizeof(float), stream);

    // 1) projections via f32 WMMA
    gemm_wmma_f32<<<(N_NODE + 15) / 16, 256, 0, stream>>>(x,  wt, xP,  N_NODE);
    gemm_wmma_f32<<<(N_EDGE + 15) / 16, 256, 0, stream>>>(xe, wt, xeP, N_EDGE);

    // 2) per-pair scores + partition functions
    pair_scores<<<N_PAIR / 8, 256, 0, stream>>>(xeP, xP, pe, pn, a, es, Ze, Zn);

    // 3) column sums (the 1e-10 background term of the dense softmax)
    colsum<<<(N_NODE + 127) / 128, 128, 0, stream>>>(xP,  N_NODE, csx);
    colsum<<<(N_EDGE + 127) / 128, 128, 0, stream>>>(xeP, N_EDGE, csxe);

    // 4) edge-side output (gather, pairs grouped by edge)
    edge_out<<<N_EDGE, 128, 0, stream>>>(xP, es, pn, Ze, csx, out_xe);

    // 5) node-side scatter + output
    node_scatter<<<N_PAIR, 128, 0, stream>>>(xeP, es, pe, pn, Sn);
    node_out<<<N_NODE, 128, 0, stream>>>(Sn, Zn, csxe, out_x);
}